// TimeAwareGATLayer_62234076119731
// MI455X (gfx1250) — compile-verified
//
#include <hip/hip_runtime.h>
#include <math.h>

// ---- problem constants (match reference) ----
#define BB 4
#define NN 1024
#define INF_DIM 128   // IN
#define FF 64
#define HH 4
#define KK 15
#define LRELU 0.2f

typedef float v2f __attribute__((ext_vector_type(2)));
typedef float v8f __attribute__((ext_vector_type(8)));

// =====================================================================
// Kernel 1: Wh = h @ W   (M=4096, K=128, C=256) with V_WMMA_F32_16X16X4_F32
// Output stored directly in (B,H,N,F) layout.
// grid = (256 m-tiles, 4), block = 128 (4 waves, one 16-col tile each)
// =====================================================================
__global__ void __launch_bounds__(128) gemm_wh(const float* __restrict__ h,
                                               const float* __restrict__ W,
                                               float* __restrict__ Wh) {
    const int lane = threadIdx.x & 31;
    const int wave = threadIdx.x >> 5;
    const int m0 = blockIdx.x * 16;                 // row tile over B*N
    const int n0 = blockIdx.y * 64 + wave * 16;     // col tile over H*F=256
    const int l16 = lane & 15;
    const int kh  = (lane >> 4) << 1;               // 0 or 2 (K striping)

    v8f acc = {};
    for (int k0 = 0; k0 < INF_DIM; k0 += 4) {
        // A fragment: lane holds A[m0+l16][k0+kh .. k0+kh+1] (contiguous 8B)
        v2f a = *(const v2f*)&h[(m0 + l16) * INF_DIM + k0 + kh];
        // B fragment: lane holds W[k0+kh][n0+l16], W[k0+kh+1][n0+l16]
        v2f b;
        b.x = W[(k0 + kh    ) * (HH * FF) + n0 + l16];
        b.y = W[(k0 + kh + 1) * (HH * FF) + n0 + l16];
        acc = __builtin_amdgcn_wmma_f32_16x16x4_f32(false, a, false, b,
                                                    (short)0, acc, false, false);
    }
    // C layout: VGPR r -> M = m0 + r + 8*(lane>=16), N = n0 + l16
    const int half = (lane >> 4) << 3;
#pragma unroll
    for (int r = 0; r < 8; ++r) {
        const int m = m0 + r + half;          // m = b*N + n
        const int c = n0 + l16;               // c = h*F + f
        const int b_ = m >> 10, n_ = m & (NN - 1);
        const int h_ = c >> 6,  f_ = c & (FF - 1);
        Wh[(((b_ * HH + h_) * NN) + n_) * FF + f_] = acc[r];
    }
}

// =====================================================================
// Kernel 2: per-row scalars src/dst/sproj/ssim.  One wave32 per (b,h,n) row.
// =====================================================================
__global__ void __launch_bounds__(256) rowstats(const float* __restrict__ Wh,
                                                const float* __restrict__ a,
                                                const float* __restrict__ wproj,
                                                const float* __restrict__ wsim,
                                                float* __restrict__ srcA,
                                                float* __restrict__ dstA,
                                                float* __restrict__ sprojA,
                                                float* __restrict__ ssimA) {
    const int row  = blockIdx.x * (blockDim.x >> 5) + (threadIdx.x >> 5);
    const int lane = threadIdx.x & 31;
    if (row >= BB * HH * NN) return;
    const int hh = (row >> 10) & (HH - 1);
    const float* r = Wh + (long)row * FF;

    float s0 = 0.f, s1 = 0.f, s2 = 0.f, s3 = 0.f;
#pragma unroll
    for (int f = lane; f < FF; f += 32) {
        const float v = r[f];
        s0 += v * a[hh * 2 * FF + f];
        s1 += v * a[hh * 2 * FF + FF + f];
        s2 += v * wproj[f];
        s3 += v * wsim[f];
    }
#pragma unroll
    for (int off = 16; off > 0; off >>= 1) {
        s0 += __shfl_xor(s0, off, 32);
        s1 += __shfl_xor(s1, off, 32);
        s2 += __shfl_xor(s2, off, 32);
        s3 += __shfl_xor(s3, off, 32);
    }
    if (lane == 0) {
        srcA[row] = s0; dstA[row] = s1; sprojA[row] = s2; ssimA[row] = s3;
    }
}

// =====================================================================
// Kernel 3: score row + dual top-k + softmax -> 30 (idx, weight) pairs/row.
// One 256-thread WG per (b,i); P/S/qb row cached in LDS, shared by 4 heads.
// =====================================================================
__global__ void __launch_bounds__(256) attn_topk(const float* __restrict__ P,
                                                 const float* __restrict__ S,
                                                 const int*   __restrict__ quad,
                                                 const float* __restrict__ qtab,
                                                 const float* __restrict__ srcA,
                                                 const float* __restrict__ dstA,
                                                 const float* __restrict__ sprojA,
                                                 const float* __restrict__ ssimA,
                                                 float* __restrict__ attn_w,
                                                 int*   __restrict__ attn_idx) {
    __shared__ float Prow[NN];
    __shared__ float Srow[NN];
    __shared__ float Qrow[NN];
    __shared__ float ebuf[NN];
    __shared__ float rval[256];
    __shared__ int   ridx[256];
    __shared__ float selv[2 * KK];
    __shared__ int   seli[2 * KK];

    const int t  = threadIdx.x;
    const int bi = blockIdx.x;             // b*N + i
    const long base = (long)bi * NN;

    for (int j = t; j < NN; j += 256) {
        Prow[j] = P[base + j];
        Srow[j] = S[base + j];
        const int q0 = quad[(base + j) * 2 + 0];
        const int q1 = quad[(base + j) * 2 + 1];
        Qrow[j] = qtab[q0 * 4 + q1];
    }
    __syncthreads();

    const int b = bi >> 10, i = bi & (NN - 1);
    for (int hh = 0; hh < HH; ++hh) {
        const int row  = ((b * HH + hh) << 10) + i;
        const int rowd = ((b * HH + hh) << 10);
        const float si = srcA[row], pi = sprojA[row], mi = ssimA[row];

        for (int j = t; j < NN; j += 256) {
            float e = si + dstA[rowd + j];
            e = e > 0.f ? e : LRELU * e;
            e += pi * Prow[j] + mi * Srow[j] + Qrow[j];
            ebuf[j] = e;
        }
        __syncthreads();

        // ---- 15 argmax passes ----
        for (int p = 0; p < KK; ++p) {
            float bv = -3.0e38f; int bj = NN;
            for (int j = t; j < NN; j += 256) {
                const float v = ebuf[j];
                if (v > bv) { bv = v; bj = j; }
            }
            rval[t] = bv; ridx[t] = bj;
            __syncthreads();
            for (int s = 128; s > 0; s >>= 1) {
                if (t < s) {
                    const float ov = rval[t + s]; const int oj = ridx[t + s];
                    if (ov > rval[t] || (ov == rval[t] && oj < ridx[t])) {
                        rval[t] = ov; ridx[t] = oj;
                    }
                }
                __syncthreads();
            }
            if (t == 0) { selv[p] = rval[0]; seli[p] = ridx[0]; ebuf[ridx[0]] = -3.0e38f; }
            __syncthreads();
        }
        // restore the claimed maxima so the min passes see true values
        if (t < KK) ebuf[seli[t]] = selv[t];
        __syncthreads();

        // ---- 15 argmin passes ----
        for (int p = KK; p < 2 * KK; ++p) {
            float bv = 3.0e38f; int bj = NN;
            for (int j = t; j < NN; j += 256) {
                const float v = ebuf[j];
                if (v < bv) { bv = v; bj = j; }
            }
            rval[t] = bv; ridx[t] = bj;
            __syncthreads();
            for (int s = 128; s > 0; s >>= 1) {
                if (t < s) {
                    const float ov = rval[t + s]; const int oj = ridx[t + s];
                    if (ov < rval[t] || (ov == rval[t] && oj < ridx[t])) {
                        rval[t] = ov; ridx[t] = oj;
                    }
                }
                __syncthreads();
            }
            if (t == 0) { selv[p] = rval[0]; seli[p] = ridx[0]; ebuf[ridx[0]] = 3.0e38f; }
            __syncthreads();
        }

        // ---- softmax over the 30 survivors (NEG entries underflow to 0) ----
        if (t == 0) {
            float mx = -3.0e38f;
            for (int p = 0; p < 2 * KK; ++p) mx = fmaxf(mx, selv[p]);
            float sum = 0.f;
            for (int p = 0; p < 2 * KK; ++p) sum += __expf(selv[p] - mx);
            const float inv = 1.f / sum;
            const long o = (long)row * (2 * KK);
            for (int p = 0; p < 2 * KK; ++p) {
                attn_w[o + p]   = __expf(selv[p] - mx) * inv;
                attn_idx[o + p] = seli[p];
            }
        }
        __syncthreads();
    }
}

// =====================================================================
// Kernel 4: sparse attn @ Wh  -> hp in (B,N,H*F) layout.
// One WG per (b,i); thread = (h,f).  Wh (4MB) is L2-resident.
// =====================================================================
__global__ void __launch_bounds__(256) gather_hp(const float* __restrict__ Wh,
                                                 const float* __restrict__ attn_w,
                                                 const int*   __restrict__ attn_idx,
                                                 float* __restrict__ hp) {
    const int t  = threadIdx.x;
    const int bi = blockIdx.x;
    const int b = bi >> 10, i = bi & (NN - 1);
    const int hh = t >> 6, f = t & (FF - 1);
    const int row = ((b * HH + hh) << 10) + i;
    const float* wp = attn_w   + (long)row * (2 * KK);
    const int*   ip = attn_idx + (long)row * (2 * KK);
    const float* WhH = Wh + ((long)(b * HH + hh) << 10) * FF;

    float acc = 0.f;
#pragma unroll
    for (int p = 0; p < 2 * KK; ++p)
        acc += wp[p] * WhH[ip[p] * FF + f];
    hp[(long)bi * (HH * FF) + hh * FF + f] = acc;
}

// =====================================================================
// Kernel 5: out = elu(hp @ Wc)  (M=4096, K=256, C=64), WMMA f32 + ELU.
// grid = 256 m-tiles, block = 128 (4 waves cover the 64 output cols)
// =====================================================================
__global__ void __launch_bounds__(128) gemm_out(const float* __restrict__ hp,
                                                const float* __restrict__ Wc,
                                                float* __restrict__ out) {
    const int lane = threadIdx.x & 31;
    const int wave = threadIdx.x >> 5;
    const int m0 = blockIdx.x * 16;
    const int n0 = wave * 16;
    const int l16 = lane & 15;
    const int kh  = (lane >> 4) << 1;

    v8f acc = {};
    for (int k0 = 0; k0 < HH * FF; k0 += 4) {
        v2f a = *(const v2f*)&hp[(m0 + l16) * (HH * FF) + k0 + kh];
        v2f b;
        b.x = Wc[(k0 + kh    ) * FF + n0 + l16];
        b.y = Wc[(k0 + kh + 1) * FF + n0 + l16];
        acc = __builtin_amdgcn_wmma_f32_16x16x4_f32(false, a, false, b,
                                                    (short)0, acc, false, false);
    }
    const int half = (lane >> 4) << 3;
#pragma unroll
    for (int r = 0; r < 8; ++r) {
        const int m = m0 + r + half;
        float v = acc[r];
        v = v > 0.f ? v : (__expf(v) - 1.f);        // ELU
        out[(long)m * FF + n0 + l16] = v;
    }
}

// =====================================================================
extern "C" void kernel_launch(void* const* d_in, const int* in_sizes, int n_in,
                              void* d_out, int out_size, void* d_ws, size_t ws_size,
                              hipStream_t stream) {
    (void)in_sizes; (void)n_in; (void)out_size; (void)ws_size;
    const float* h     = (const float*)d_in[0];
    const float* P     = (const float*)d_in[1];
    const float* S     = (const float*)d_in[2];
    const int*   quad  = (const int*)  d_in[3];
    const float* W     = (const float*)d_in[4];
    const float* a     = (const float*)d_in[5];
    const float* wproj = (const float*)d_in[6];
    const float* wsim  = (const float*)d_in[7];
    const float* qtab  = (const float*)d_in[8];
    const float* Wc    = (const float*)d_in[9];
    float* out = (float*)d_out;

    // workspace layout (floats): Wh | src | dst | sproj | ssim | hp | attn_w | attn_idx
    float* ws     = (float*)d_ws;
    float* Wh     = ws;                       // B*H*N*F  = 1,048,576
    float* srcA   = Wh     + BB * HH * NN * FF;
    float* dstA   = srcA   + BB * HH * NN;    // 16384 each
    float* sprojA = dstA   + BB * HH * NN;
    float* ssimA  = sprojA + BB * HH * NN;
    float* hp     = ssimA  + BB * HH * NN;    // B*N*H*F = 1,048,576
    float* attn_w = hp     + BB * NN * HH * FF;
    int*   attn_idx = (int*)(attn_w + (size_t)BB * HH * NN * 2 * KK);

    dim3 g1((BB * NN) / 16, (HH * FF) / 64);
    gemm_wh<<<g1, 128, 0, stream>>>(h, W, Wh);

    rowstats<<<(BB * HH * NN) / 8, 256, 0, stream>>>(Wh, a, wproj, wsim,
                                                     srcA, dstA, sprojA, ssimA);

    attn_topk<<<BB * NN, 256, 0, stream>>>(P, S, quad, qtab,
                                           srcA, dstA, sprojA, ssimA,
                                           attn_w, attn_idx);

    gather_hp<<<BB * NN, 256, 0, stream>>>(Wh, attn_w, attn_idx, hp);

    gemm_out<<<(BB * NN) / 16, 128, 0, stream>>>(hp, Wc, out);
}